// BaselineNCA_23802708754927
// MI455X (gfx1250) — compile-verified
//
#include <hip/hip_runtime.h>

typedef __attribute__((ext_vector_type(2))) float v2f;
typedef __attribute__((ext_vector_type(8))) float v8f;

#define B_    16
#define CH    4
#define H_    512
#define W_    512
#define HN    6
#define ROWS  4
#define LSTR  520              // padded row: [3]=left halo, [4..515]=cols, [516]=right halo
#define MAINO 4                // main columns start at float index 4 (16B aligned)
#define NSEG  (CH * 6)         // 4 channels x 6 staged rows

// wave32 cross-half swap: ds_swizzle group-of-32, xor_mask=0x10, and_mask=0x1f
__device__ __forceinline__ float swapx16(float v) {
  return __int_as_float(__builtin_amdgcn_ds_swizzle(__float_as_int(v), 0x401f));
}

// CDNA5 async global->LDS copies (tracked with ASYNCcnt).
// VDST = per-lane LDS byte address, VADDR = per-lane 64-bit global address.
__device__ __forceinline__ void async_b128(unsigned lds_off, const float* g) {
  asm volatile("global_load_async_to_lds_b128 %0, %1, off"
               :: "v"(lds_off), "v"(g) : "memory");
}
__device__ __forceinline__ void async_b32(unsigned lds_off, const float* g) {
  asm volatile("global_load_async_to_lds_b32 %0, %1, off"
               :: "v"(lds_off), "v"(g) : "memory");
}
__device__ __forceinline__ void wait_async0() {
  asm volatile("s_wait_asynccnt 0x0" ::: "memory");
}

__global__ __launch_bounds__(256) void nca_fused(
    const float* __restrict__ x,
    const float* __restrict__ w1w, const float* __restrict__ w1b,
    const float* __restrict__ w2w, const float* __restrict__ w2b,
    const float* __restrict__ w3w, const float* __restrict__ w3b,
    float* __restrict__ out)
{
  __shared__ float xs[NSEG * LSTR];

  const int  tid  = threadIdx.x;
  const int  lane = tid & 31;
  const int  m    = lane & 15;
  const bool hi   = lane >= 16;
  const int  wave = tid >> 5;
  const int  blk  = blockIdx.x;
  const int  b    = blk >> 7;            // 128 row-blocks per batch image
  const int  h0   = (blk & 127) * ROWS;

  // ---- async-stage x rows [h0-1 .. h0+4] x 4 channels into LDS ----
  // main columns: 24 segments x 128 float4 = 3072 b128 transfers (12/thread)
#pragma unroll
  for (int it = 0; it < (NSEG * (W_ / 4)) / 256; ++it) {
    const int i    = tid + it * 256;
    const int seg  = i >> 7;             // c*6 + r
    const int col4 = (i & 127) << 2;
    const int c    = seg / 6;
    const int r    = seg - c * 6;
    const int gr   = (h0 - 1 + r) & (H_ - 1);
    const unsigned lds_off =
        (unsigned)(uintptr_t)(xs + seg * LSTR + MAINO + col4);
    async_b128(lds_off, x + ((b * CH + c) * H_ + gr) * W_ + col4);
  }
  // wrap halos: 2 elements per segment (col -1 <- 511, col 512 <- 0)
  if (tid < 2 * NSEG) {
    const int seg  = tid >> 1;
    const int side = tid & 1;            // 0=left halo, 1=right halo
    const int c    = seg / 6;
    const int r    = seg - c * 6;
    const int gr   = (h0 - 1 + r) & (H_ - 1);
    const int gc   = side ? 0 : (W_ - 1);
    const unsigned lds_off =
        (unsigned)(uintptr_t)(xs + seg * LSTR + (side ? MAINO + W_ : MAINO - 1));
    async_b32(lds_off, x + ((b * CH + c) * H_ + gr) * W_ + gc);
  }
  wait_async0();
  __syncthreads();

  // ---- per-lane constants ----
  // A-operand: W1 (6x16) in rows 0..5, duplicated into rows 8..13 so that the
  // C-matrix upper-lane half (M=8..15) carries the same hidden vector.
  const int er = (m < HN) ? m : ((m >= 8 && m < 8 + HN) ? (m - 8) : -1);
  v2f a[4];
#pragma unroll
  for (int k = 0; k < 4; ++k) {
    const int kk = 4 * k + (hi ? 2 : 0);
    float ax = 0.f, ay = 0.f;
    if (er >= 0) { ax = w1w[er * 16 + kk]; ay = w1w[er * 16 + kk + 1]; }
    a[k].x = ax; a[k].y = ay;
  }
  v8f cinit;
#pragma unroll
  for (int r = 0; r < 8; ++r) cinit[r] = (r < HN) ? w1b[r] : 0.f;

  // layer-2 weights: lower half of wave -> w2 (tanh), upper half -> w3 (sigmoid)
  float ws[4][HN], bs[4];
#pragma unroll
  for (int c = 0; c < 4; ++c) {
    bs[c] = hi ? w3b[c] : w2b[c];
#pragma unroll
    for (int r = 0; r < HN; ++r)
      ws[c][r] = hi ? w3w[c * HN + r] : w2w[c * HN + r];
  }

  // ---- each wave: 256 pixels = 8 passes x 32 pixels (two 16-px WMMA tiles) ----
#pragma unroll 1
  for (int pass = 0; pass < 8; ++pass) {
    const int flat = wave * 256 + pass * 32;   // 0..2047 within 4x512 tile
    const int row  = flat >> 9;
    const int col0 = flat & (W_ - 1);
    const int h    = h0 + row;
    const int lr   = row + 1;                  // LDS row of conv center
    const int px   = col0 + lane;              // this lane's conv pixel

    // conv: all 4 filter responses for this lane's pixel, all 4 channels
    float id[CH], sx[CH], sy[CH], lp[CH];
#pragma unroll
    for (int c = 0; c < CH; ++c) {
      // [0]=col px-1, [1]=px, [2]=px+1  (halo-adjusted base)
      const float* rt = xs + (c * 6 + lr - 1) * LSTR + (MAINO - 1) + px;
      const float* rm = rt + LSTR;
      const float* rb = rm + LSTR;
      const float tl = rt[0], tm = rt[1], tr = rt[2];
      const float ml = rm[0], mm = rm[1], mr = rm[2];
      const float bl = rb[0], bm = rb[1], br = rb[2];
      id[c] = mm;
      sx[c] = ((tr - tl) + (br - bl)) + 2.f * (mr - ml);
      const float u1 = bl + br, u2 = tl + tr;
      sy[c] = (u1 - u2) + 2.f * (bm - tm);
      lp[c] = fmaf(-12.f, mm, (u1 + u2) + 2.f * ((tm + bm) + (ml + mr)));
    }

    // Build B operands (4x16 f32, K = 4*c + filter):
    // lanes 0-15 hold K%4 in {0,1} (identity, sobel_x), lanes 16-31 hold {2,3}.
    v2f bA[4], bB[4];
#pragma unroll
    for (int c = 0; c < CH; ++c) {
      const float id_s = swapx16(id[c]);
      const float sx_s = swapx16(sx[c]);
      const float sy_s = swapx16(sy[c]);
      const float lp_s = swapx16(lp[c]);
      bA[c].x = hi ? sy_s  : id[c];
      bA[c].y = hi ? lp_s  : sx[c];
      bB[c].x = hi ? sy[c] : id_s;
      bB[c].y = hi ? lp[c] : sx_s;
    }

    // hid = relu(W1 * Y + b1) via 4 chained fp32 WMMAs per tile
    v8f cA = cinit, cB = cinit;
#pragma unroll
    for (int k = 0; k < 4; ++k) {
      cA = __builtin_amdgcn_wmma_f32_16x16x4_f32(false, a[k], false, bA[k],
                                                 (short)0, cA, false, false);
      cB = __builtin_amdgcn_wmma_f32_16x16x4_f32(false, a[k], false, bB[k],
                                                 (short)0, cB, false, false);
    }

    const v8f acc2[2] = {cA, cB};
#pragma unroll
    for (int t = 0; t < 2; ++t) {
      float hv[HN];
#pragma unroll
      for (int r = 0; r < HN; ++r) hv[r] = fmaxf(acc2[t][r], 0.f);

      // lower half: val = tanh(z) ; upper half: val = sigmoid(z)
      float val[4], vs[4];
#pragma unroll
      for (int c = 0; c < 4; ++c) {
        float z = bs[c];
#pragma unroll
        for (int r = 0; r < HN; ++r) z = fmaf(ws[c][r], hv[r], z);
        z = fminf(fmaxf(z, -20.f), 20.f);
        const float e  = __expf(hi ? -z : (z + z));
        const float rc = __builtin_amdgcn_rcpf(1.f + e);
        val[c] = hi ? rc : (e - 1.f) * rc;     // tanh=(e2z-1)/(e2z+1), sig=1/(1+e-z)
      }
#pragma unroll
      for (int c = 0; c < 4; ++c) vs[c] = swapx16(val[c]);

      const int p = col0 + t * 16 + m;
#pragma unroll
      for (int j = 0; j < 2; ++j) {
        const int   ch = (hi ? 2 : 0) + j;     // lanes 0-15: ch 0,1 ; 16-31: ch 2,3
        const float u  = hi ? vs[2 + j]  : val[j];       // y_upd
        const float g  = hi ? val[2 + j] : vs[j];        // gate
        const float xc = xs[(ch * 6 + lr) * LSTR + MAINO + p];
        out[((b * CH + ch) * H_ + h) * W_ + p] = fmaf(g, xc - u, u);
      }
    }
  }
}

extern "C" void kernel_launch(void* const* d_in, const int* in_sizes, int n_in,
                              void* d_out, int out_size, void* d_ws, size_t ws_size,
                              hipStream_t stream) {
  const float* x   = (const float*)d_in[0];
  // d_in[1] = filters; fixed identity/sobel_x/sobel_y/laplacian, baked into kernel
  const float* w1w = (const float*)d_in[2];
  const float* w1b = (const float*)d_in[3];
  const float* w2w = (const float*)d_in[4];
  const float* w2b = (const float*)d_in[5];
  const float* w3w = (const float*)d_in[6];
  const float* w3b = (const float*)d_in[7];
  float* out = (float*)d_out;

  dim3 grid(B_ * (H_ / ROWS));   // 16 * 128 = 2048 workgroups
  nca_fused<<<grid, 256, 0, stream>>>(x, w1w, w1b, w2w, w2b, w3w, w3b, out);
}